// VectorQuantizer_63728724738241
// MI455X (gfx1250) — compile-verified
//
#include <hip/hip_runtime.h>
#include <math.h>

typedef float v2f __attribute__((ext_vector_type(2)));
typedef float v8f __attribute__((ext_vector_type(8)));

#define K_CODES   512
#define DIM       64
#define NROWS     65536      // B*H*W
#define SPATIAL   1024       // H*W
#define CB_STRIDE 65         // padded LDS stride (bank-conflict free)

// output layout (floats), concatenated in reference return order
#define OFF_QUANT   0
#define OFF_LOSS    4194304
#define OFF_PERP    4194305
#define OFF_IDX     4194306   // encoding_indice  [64,1024]
#define OFF_IDXPROG 4259842   // index_program    [64,512]
#define OFF_SOFT    4292610   // softmax_histogram[64,512]

__global__ void vq_zero_kernel(float* out, float* ws) {
  int i = blockIdx.x * blockDim.x + threadIdx.x;
  if (i < 32768) { out[OFF_IDXPROG + i] = 0.f; out[OFF_SOFT + i] = 0.f; }
  if (i < 514) ws[i] = 0.f;
  if (i == 0) { out[OFF_LOSS] = 0.f; out[OFF_PERP] = 0.f; }
}

__launch_bounds__(256)
__global__ void vq_main_kernel(const float* __restrict__ x,
                               const float* __restrict__ cb,
                               float* __restrict__ out,
                               float* __restrict__ ws) {
  extern __shared__ float smem[];                 // 512*65 codebook + 512 norms
  float* scb = smem;
  float* scn = smem + K_CODES * CB_STRIDE;

  const int lane  = threadIdx.x & 31;
  const int wave  = threadIdx.x >> 5;
  const int tile  = blockIdx.x * 8 + wave;        // 512 blocks * 8 waves = 4096 tiles
  const int row0  = tile * 16;
  const int b     = row0 >> 10;                   // batch (tile never straddles a batch)
  const int sbase = row0 & 1023;                  // spatial base within batch
  const int ml    = lane & 15;
  const int half  = lane >> 4;

  const float* xb = x + (size_t)b * (DIM * SPATIAL);  // NCHW: x[b, d, s]

  // prefetch my x tile while the codebook streams into LDS (global_prefetch_b8)
  #pragma unroll
  for (int s = 0; s < 16; s += 4)
    __builtin_prefetch(&xb[(size_t)(4 * s + 2 * half) * SPATIAL + sbase + ml], 0, 0);

  // cooperative codebook load into LDS (134KB resident; 320KB/WGP available)
  for (int t = threadIdx.x; t < K_CODES * DIM; t += 256) {
    int k = t >> 6, d = t & 63;
    scb[k * CB_STRIDE + d] = cb[t];
  }
  __syncthreads();
  for (int k = threadIdx.x; k < K_CODES; k += 256) {
    float s = 0.f;
    #pragma unroll
    for (int d = 0; d < DIM; ++d) { float v = scb[k * CB_STRIDE + d]; s += v * v; }
    scn[k] = s;
  }
  __syncthreads();

  // A fragments: 16 k-steps of 16x4 fp32. Lane<16: K=4s,4s+1; lane>=16: K=4s+2,4s+3.
  v2f afrag[16];
  #pragma unroll
  for (int s = 0; s < 16; ++s) {
    int d0 = 4 * s + 2 * half;
    afrag[s].x = xb[(size_t)(d0    ) * SPATIAL + sbase + ml];
    afrag[s].y = xb[(size_t)(d0 + 1) * SPATIAL + sbase + ml];
  }

  // ||x||^2 per row: partial over my half's K-slices, fold across halves
  float part = 0.f;
  #pragma unroll
  for (int s = 0; s < 16; ++s) part += afrag[s].x * afrag[s].x + afrag[s].y * afrag[s].y;
  part += __shfl_xor(part, 16, 32);
  float xn[8];
  #pragma unroll
  for (int i = 0; i < 8; ++i) xn[i] = __shfl(part, i + 8 * half, 32);  // row = i + 8*half

  // lane-local state per slot-row: running min, argmin, online softmax sum
  float rm[8], rs[8]; int ri[8];
  #pragma unroll
  for (int i = 0; i < 8; ++i) { rm[i] = 3.4e38f; rs[i] = 0.f; ri[i] = 0; }

  // ---- pass 1: WMMA dots -> lane-local argmin + online log-sum-exp (no shuffles) ----
  for (int c = 0; c < 32; ++c) {
    const int col = c * 16 + ml;
    v8f acc = {};
    #pragma unroll
    for (int s = 0; s < 16; ++s) {
      int d0 = 4 * s + 2 * half;
      v2f bfrag;
      bfrag.x = scb[col * CB_STRIDE + d0];
      bfrag.y = scb[col * CB_STRIDE + d0 + 1];
      acc = __builtin_amdgcn_wmma_f32_16x16x4_f32(false, afrag[s], false, bfrag,
                                                  (short)0, acc, false, false);
    }
    const float cn = scn[col];
    #pragma unroll
    for (int i = 0; i < 8; ++i) {
      float dst  = fmaf(-2.f, acc[i], xn[i] + cn);
      int   take = (int)(dst < rm[i]);                 // strict < keeps lowest col
      float nm   = fminf(rm[i], dst);
      ri[i] = take ? col : ri[i];
      rs[i] = rs[i] * __expf(nm - rm[i]) + __expf(nm - dst);
      rm[i] = nm;
    }
  }

  // one cross-lane combine per tile: merge (m, idx, s) across the 16-lane half
  #pragma unroll
  for (int i = 0; i < 8; ++i) {
    float m = rm[i], s = rs[i]; int id = ri[i];
    #pragma unroll
    for (int msk = 1; msk <= 8; msk <<= 1) {
      float m2 = __shfl_xor(m, msk, 32);
      float s2 = __shfl_xor(s, msk, 32);
      int   i2 = __shfl_xor(id, msk, 32);
      float nm = fminf(m, m2);
      s = s * __expf(nm - m) + s2 * __expf(nm - m2);
      int take = (int)(m2 < m) | ((int)(m2 == m) & (int)(i2 < id));
      id = take ? i2 : id;
      m = nm;
    }
    rm[i] = m; rs[i] = 1.f / s; ri[i] = id;   // store reciprocal for pass 2
  }

  // per-row outputs: indices + code-usage histograms
  if (ml == 0) {
    #pragma unroll
    for (int i = 0; i < 8; ++i) {
      int row = row0 + i + 8 * half;
      out[OFF_IDX + row] = (float)ri[i];
      atomicAdd(&out[OFF_IDXPROG + b * K_CODES + ri[i]], 1.f);
      atomicAdd(&ws[ri[i]], 1.f);
    }
  }

  // ---- pass 2: recompute dots, scatter normalized softmax into per-batch hist ----
  for (int c = 0; c < 32; ++c) {
    const int col = c * 16 + ml;
    v8f acc = {};
    #pragma unroll
    for (int s = 0; s < 16; ++s) {
      int d0 = 4 * s + 2 * half;
      v2f bfrag;
      bfrag.x = scb[col * CB_STRIDE + d0];
      bfrag.y = scb[col * CB_STRIDE + d0 + 1];
      acc = __builtin_amdgcn_wmma_f32_16x16x4_f32(false, afrag[s], false, bfrag,
                                                  (short)0, acc, false, false);
    }
    const float cn = scn[col];
    float h = 0.f;
    #pragma unroll
    for (int i = 0; i < 8; ++i) {
      float dst = fmaf(-2.f, acc[i], xn[i] + cn);
      h += __expf(rm[i] - dst) * rs[i];
    }
    h += __shfl_xor(h, 16, 32);   // fold rows 8..15 onto rows 0..7 (same col)
    if (half == 0) atomicAdd(&out[OFF_SOFT + b * K_CODES + col], h);
  }
}

__launch_bounds__(256)
__global__ void vq_quant_loss_kernel(const float* __restrict__ x,
                                     const float* __restrict__ cb,
                                     float* __restrict__ out,
                                     float* __restrict__ ws) {
  __shared__ float red[256];
  int o = blockIdx.x * 256 + threadIdx.x;   // NCHW flat index < 4194304
  int b = o >> 16;
  int r = o & 65535;
  int d = r >> 10;
  int s = r & 1023;
  int k = (int)out[OFF_IDX + b * SPATIAL + s];
  float q = cb[k * DIM + d];
  out[OFF_QUANT + o] = q;                   // straight-through value == quantized
  float diff = q - x[o];
  red[threadIdx.x] = diff * diff;
  __syncthreads();
  for (int st = 128; st > 0; st >>= 1) {
    if (threadIdx.x < st) red[threadIdx.x] += red[threadIdx.x + st];
    __syncthreads();
  }
  if (threadIdx.x == 0) atomicAdd(&ws[512], red[0]);
}

__global__ void vq_finalize_kernel(float* out, const float* ws) {
  __shared__ float red[512];
  int k = threadIdx.x;
  float p = ws[k] * (1.f / 65536.f);
  red[k] = -p * __logf(p + 1e-10f);
  __syncthreads();
  for (int st = 256; st > 0; st >>= 1) {
    if (k < st) red[k] += red[k + st];
    __syncthreads();
  }
  if (k == 0) {
    out[OFF_PERP] = __expf(red[0]);
    out[OFF_LOSS] = ws[512] * (1.25f / 4194304.f);  // q_loss + 0.25*e_loss (equal values)
  }
}

extern "C" void kernel_launch(void* const* d_in, const int* in_sizes, int n_in,
                              void* d_out, int out_size, void* d_ws, size_t ws_size,
                              hipStream_t stream) {
  const float* x  = (const float*)d_in[0];   // [64,64,32,32]
  const float* cb = (const float*)d_in[1];   // [512,64]
  float* out = (float*)d_out;
  float* ws  = (float*)d_ws;

  vq_zero_kernel<<<128, 256, 0, stream>>>(out, ws);

  size_t lds_bytes = (size_t)(K_CODES * CB_STRIDE + K_CODES) * sizeof(float); // ~134 KB
  vq_main_kernel<<<512, 256, lds_bytes, stream>>>(x, cb, out, ws);

  vq_quant_loss_kernel<<<4194304 / 256, 256, 0, stream>>>(x, cb, out, ws);
  vq_finalize_kernel<<<1, 512, 0, stream>>>(out, ws);
}